// Attention_28870770164037
// MI455X (gfx1250) — compile-verified
//
#include <hip/hip_runtime.h>
#include <hip/hip_bf16.h>

// ---------------- problem constants ----------------
#define Bb 8
#define Ss 197
#define Ee 384
#define Hh 6
#define Dd 64
#define Nn (Bb*Ss)         // 1576
#define NT ((size_t)Ee*Nn) // 605184 elements per projection output
#define SP 208             // S padded to 13*16
#define NP 1584            // N padded to 99*16

typedef __attribute__((ext_vector_type(2))) float v2f;
typedef __attribute__((ext_vector_type(8))) float v8f;

// =====================================================================
// Kernel 1: AdderNet 1x1 "conv": Y[co,n] = -sum_ci |X[ci,n] - W[co,ci]|
// X is x flat-reinterpreted as [E][N]. Tiled 64(co) x 64(n), 4x4/thread.
// Also emits per-block (sum, sumsq) partials for the global LayerNorm.
// =====================================================================
__global__ void adder_kernel(const float* __restrict__ x,
                             const float* __restrict__ wq,
                             const float* __restrict__ wk,
                             const float* __restrict__ wv,
                             float* __restrict__ Y,
                             float* __restrict__ partial)
{
    __shared__ float Ws[64][17];
    __shared__ float Xs[16][65];
    __shared__ float rs1[256], rs2[256];

    const int tid = threadIdx.x;
    const int bz  = blockIdx.z;
    const float* __restrict__ w = (bz == 0) ? wq : (bz == 1) ? wk : wv;
    float* __restrict__ Yp = Y + (size_t)bz * NT;

    const int nBase  = blockIdx.x * 64;
    const int coBase = blockIdx.y * 64;
    const int tx = tid & 15, ty = tid >> 4;

    float acc[4][4];
#pragma unroll
    for (int i = 0; i < 4; i++)
#pragma unroll
        for (int j = 0; j < 4; j++) acc[i][j] = 0.f;

    for (int cc = 0; cc < Ee; cc += 16) {
#pragma unroll
        for (int q = 0; q < 4; q++) {               // W tile 64x16
            int l = tid + 256 * q;
            int r = l >> 4, c = l & 15;
            Ws[r][c] = w[(size_t)(coBase + r) * Ee + cc + c];
        }
#pragma unroll
        for (int q = 0; q < 4; q++) {               // X tile 16x64
            int l = tid + 256 * q;
            int r = l >> 6, c = l & 63;
            int n = nBase + c;
            Xs[r][c] = (n < Nn) ? x[(size_t)(cc + r) * Nn + n] : 0.f;
        }
        __syncthreads();
#pragma unroll
        for (int kk = 0; kk < 16; kk++) {
            float wv4[4], xv4[4];
#pragma unroll
            for (int i = 0; i < 4; i++) wv4[i] = Ws[ty + 16 * i][kk];
#pragma unroll
            for (int j = 0; j < 4; j++) xv4[j] = Xs[kk][tx + 16 * j];
#pragma unroll
            for (int i = 0; i < 4; i++)
#pragma unroll
                for (int j = 0; j < 4; j++)
                    acc[i][j] -= fabsf(xv4[j] - wv4[i]);  // abs folds to |src| modifier
        }
        __syncthreads();
    }

    float s1 = 0.f, s2 = 0.f;
#pragma unroll
    for (int i = 0; i < 4; i++) {
        int co = coBase + ty + 16 * i;
#pragma unroll
        for (int j = 0; j < 4; j++) {
            int n = nBase + tx + 16 * j;
            if (n < Nn) {
                float v = acc[i][j];
                Yp[(size_t)co * Nn + n] = v;
                s1 += v; s2 += v * v;
            }
        }
    }
    rs1[tid] = s1; rs2[tid] = s2; __syncthreads();
    for (int off = 128; off > 0; off >>= 1) {
        if (tid < off) { rs1[tid] += rs1[tid + off]; rs2[tid] += rs2[tid + off]; }
        __syncthreads();
    }
    if (tid == 0) {
        int pidx = bz * 150 + blockIdx.y * 25 + blockIdx.x;
        partial[2 * pidx]     = rs1[0];
        partial[2 * pidx + 1] = rs2[0];
    }
}

// =====================================================================
// Kernel 2: fold 150 partials per projection -> (mean, inv_std)
// =====================================================================
__global__ void stats_kernel(const float* __restrict__ partial, float* __restrict__ stats)
{
    __shared__ float rs1[256], rs2[256];
    int z = blockIdx.x, tid = threadIdx.x;
    float s1 = 0.f, s2 = 0.f;
    for (int i = tid; i < 150; i += 256) {
        s1 += partial[2 * (z * 150 + i)];
        s2 += partial[2 * (z * 150 + i) + 1];
    }
    rs1[tid] = s1; rs2[tid] = s2; __syncthreads();
    for (int off = 128; off > 0; off >>= 1) {
        if (tid < off) { rs1[tid] += rs1[tid + off]; rs2[tid] += rs2[tid + off]; }
        __syncthreads();
    }
    if (tid == 0) {
        float mean = rs1[0] / (float)NT;
        float var  = rs2[0] / (float)NT - mean * mean;
        stats[2 * z]     = mean;
        stats[2 * z + 1] = rsqrtf(var + 1e-5f);
    }
}

// =====================================================================
// Kernel 3: apply global LN (per-element affine, same flat index as Y)
// and permute into padded q/k/v: [3][48][SP][64], padded rows = 0.
// =====================================================================
__global__ void qkv_build_kernel(const float* __restrict__ Y,
                                 const float* __restrict__ stats,
                                 const float* __restrict__ qlw, const float* __restrict__ qlb,
                                 const float* __restrict__ klw, const float* __restrict__ klb,
                                 const float* __restrict__ vlw, const float* __restrict__ vlb,
                                 float* __restrict__ qkv)
{
    const int z = blockIdx.z;
    const float* lw = (z == 0) ? qlw : (z == 1) ? klw : vlw;
    const float* lb = (z == 0) ? qlb : (z == 1) ? klb : vlb;
    const float* Yp = Y + (size_t)z * NT;
    const float mean = stats[2 * z], istd = stats[2 * z + 1];
    const int bh = blockIdx.y;
    const int b = bh / Hh, h = bh % Hh;
    float* out = qkv + ((size_t)z * 48 + bh) * SP * Dd;
    const int tid = threadIdx.x;
#pragma unroll
    for (int q = 0; q < 4; q++) {
        int l = tid + 256 * q;
        int sLoc = l >> 6, d = l & 63;
        int s = blockIdx.x * 16 + sLoc;      // 0..207
        float v = 0.f;
        if (s < Ss) {
            int n = b * Ss + s;
            size_t idx = (size_t)(h * Dd + d) * Nn + n;
            v = (Yp[idx] - mean) * istd * lw[idx] + lb[idx];
        }
        out[s * Dd + d] = v;
    }
}

// =====================================================================
// Kernel 4: transpose fc_w [E][E] -> fcwT[k][co] so FC B-fragment loads
// are lane-coalesced.
// =====================================================================
__global__ void wt_kernel(const float* __restrict__ fcw, float* __restrict__ fcwT)
{
    __shared__ float t[32][33];
    const int bx = blockIdx.x * 32, by = blockIdx.y * 32;
    const int tx = threadIdx.x & 31, ty = threadIdx.x >> 5;   // 256 thr = 32x8
#pragma unroll
    for (int i = ty; i < 32; i += 8)
        t[i][tx] = fcw[(size_t)(by + i) * Ee + bx + tx];
    __syncthreads();
#pragma unroll
    for (int i = ty; i < 32; i += 8)
        fcwT[(size_t)(bx + i) * Ee + by + tx] = t[tx][i];
}

// =====================================================================
// Kernel 5: attention per (b,h), 16 query rows per block, 4 waves.
// WMMA fragments preloaded in register batches so ds_loads pipeline
// ahead of the v_wmma chain instead of s_wait_dscnt 0 per step.
// =====================================================================
__global__ void attn_kernel(const float* __restrict__ qkv, float* __restrict__ o_raw)
{
    extern __shared__ float smem[];
    float* Ks = smem;                  // SP*Dd
    float* Vs = Ks + SP * Dd;          // SP*Dd
    float* Qs = Vs + SP * Dd;          // 16*Dd
    float* Lg = Qs + 16 * Dd;          // 16*SP
    float* red    = Lg + 16 * SP;      // 128
    float* rowmax = red + 128;         // 16
    float* rowsum = rowmax + 16;       // 16

    const int tid = threadIdx.x;
    const int bh = blockIdx.y;
    const int b = bh / Hh, h = bh % Hh;
    const int rowTile = blockIdx.x;

    const float4* qsrc = (const float4*)(qkv + (size_t)bh * SP * Dd + (size_t)rowTile * 16 * Dd);
    const float4* ksrc = (const float4*)(qkv + (size_t)(48 + bh) * SP * Dd);
    const float4* vsrc = (const float4*)(qkv + (size_t)(96 + bh) * SP * Dd);
    float4* kd = (float4*)Ks; float4* vd = (float4*)Vs; float4* qd = (float4*)Qs;
    for (int i = tid; i < SP * Dd / 4; i += 128) { kd[i] = ksrc[i]; vd[i] = vsrc[i]; }
    for (int i = tid; i < 16 * Dd / 4; i += 128) { qd[i] = qsrc[i]; }
    __syncthreads();

    const int wave = tid >> 5;
    const int lane = tid & 31;
    const int l16  = lane & 15;
    const int hi   = lane >> 4;
    const float scale = rsqrtf(2.0f * (float)Dd * (1.0f - 2.0f / 3.14159265358979323846f));

    // ---- phase 1: logits = Q(16x64) * K^T, column tiles round-robin per wave
    v2f afr[16];
#pragma unroll
    for (int t = 0; t < 16; t++) {                 // Q fragments: load once
        afr[t].x = Qs[l16 * Dd + t * 4 + 2 * hi];
        afr[t].y = Qs[l16 * Dd + t * 4 + 2 * hi + 1];
    }
    for (int j = wave; j < 13; j += 4) {
        v2f bfr[16];
#pragma unroll
        for (int t = 0; t < 16; t++) {             // batch-load all B frags
            bfr[t].x = Ks[(j * 16 + l16) * Dd + t * 4 + 2 * hi];
            bfr[t].y = Ks[(j * 16 + l16) * Dd + t * 4 + 2 * hi + 1];
        }
        v8f c = {};
#pragma unroll
        for (int t = 0; t < 16; t++)
            c = __builtin_amdgcn_wmma_f32_16x16x4_f32(false, afr[t], false, bfr[t],
                                                      (short)0, c, false, false);
#pragma unroll
        for (int r = 0; r < 8; r++) {
            int row = r + hi * 8;
            int col = j * 16 + l16;
            float v = c[r] * scale;
            if (col >= Ss) v = -1e30f;             // mask padded K columns
            Lg[row * SP + col] = v;
        }
    }
    __syncthreads();

    // ---- phase 2: row softmax over 208 cols (8 threads/row) + identity add
    {
        int row = tid & 15, sub = tid >> 4;
        float m = -1e30f;
        for (int cI = sub; cI < SP; cI += 8) m = fmaxf(m, Lg[row * SP + cI]);
        red[row * 8 + sub] = m; __syncthreads();
        if (sub == 0) {
            float mm = red[row * 8];
            for (int i = 1; i < 8; i++) mm = fmaxf(mm, red[row * 8 + i]);
            rowmax[row] = mm;
        }
        __syncthreads();
        float mm = rowmax[row], ss = 0.f;
        for (int cI = sub; cI < SP; cI += 8) {
            float ev = expf(Lg[row * SP + cI] - mm);
            Lg[row * SP + cI] = ev;
            ss += ev;
        }
        red[row * 8 + sub] = ss; __syncthreads();
        if (sub == 0) {
            float t = 0.f;
            for (int i = 0; i < 8; i++) t += red[row * 8 + i];
            rowsum[row] = t;
        }
        __syncthreads();
        float inv = 1.0f / rowsum[row];
        int grow = rowTile * 16 + row;
        for (int cI = sub; cI < SP; cI += 8) {
            float v = Lg[row * SP + cI] * inv;
            if (cI == grow) v += 1.0f;             // + eye(S)
            Lg[row * SP + cI] = v;
        }
        __syncthreads();
    }

    // ---- phase 3: O(16x64) = Att(16x208) * V(208x64), one d-tile per wave,
    //      K-loop chunked 13-deep so loads pipeline ahead of WMMA.
    {
        const int dt = wave;
        v8f c = {};
        for (int kc = 0; kc < SP; kc += 52) {      // 4 chunks x 13 k-steps
            v2f af3[13], bf3[13];
#pragma unroll
            for (int t = 0; t < 13; t++) {
                int k = kc + t * 4;
                af3[t].x = Lg[l16 * SP + k + 2 * hi];
                af3[t].y = Lg[l16 * SP + k + 2 * hi + 1];
                bf3[t].x = Vs[(k + 2 * hi) * Dd + dt * 16 + l16];
                bf3[t].y = Vs[(k + 2 * hi + 1) * Dd + dt * 16 + l16];
            }
#pragma unroll
            for (int t = 0; t < 13; t++)
                c = __builtin_amdgcn_wmma_f32_16x16x4_f32(false, af3[t], false, bf3[t],
                                                          (short)0, c, false, false);
        }
#pragma unroll
        for (int r = 0; r < 8; r++) {
            int row = r + hi * 8;
            int s = rowTile * 16 + row;
            if (s < Ss) {
                int n = b * Ss + s;
                int e = h * Dd + dt * 16 + l16;
                o_raw[(size_t)n * Ee + e] = c[r];
            }
        }
    }
}

// =====================================================================
// Kernel 6: per-row LayerNorm over E=384 (+affine); pad rows -> 0
// =====================================================================
__global__ void oln_kernel(const float* __restrict__ o_raw,
                           const float* __restrict__ olw,
                           const float* __restrict__ olb,
                           float* __restrict__ o_ln)
{
    __shared__ float red[128];
    const int n = blockIdx.x;    // 0..NP-1
    const int tid = threadIdx.x;
    if (n >= Nn) {
        for (int e = tid; e < Ee; e += 128) o_ln[(size_t)n * Ee + e] = 0.f;
        return;
    }
    float v0 = o_raw[(size_t)n * Ee + tid];
    float v1 = o_raw[(size_t)n * Ee + tid + 128];
    float v2 = o_raw[(size_t)n * Ee + tid + 256];
    red[tid] = v0 + v1 + v2; __syncthreads();
    for (int off = 64; off > 0; off >>= 1) { if (tid < off) red[tid] += red[tid + off]; __syncthreads(); }
    float mu = red[0] / (float)Ee; __syncthreads();
    float d0 = v0 - mu, d1 = v1 - mu, d2 = v2 - mu;
    red[tid] = d0 * d0 + d1 * d1 + d2 * d2; __syncthreads();
    for (int off = 64; off > 0; off >>= 1) { if (tid < off) red[tid] += red[tid + off]; __syncthreads(); }
    float rs = rsqrtf(red[0] / (float)Ee + 1e-5f);
    o_ln[(size_t)n * Ee + tid]       = d0 * rs * olw[tid]       + olb[tid];
    o_ln[(size_t)n * Ee + tid + 128] = d1 * rs * olw[tid + 128] + olb[tid + 128];
    o_ln[(size_t)n * Ee + tid + 256] = d2 * rs * olw[tid + 256] + olb[tid + 256];
}

// =====================================================================
// Kernel 7: out = o_ln @ fc_w^T + fc_b via WMMA fp32.
// A-tile in LDS; B from pre-transposed fcwT (coalesced); K chunked 12-deep.
// =====================================================================
__global__ void fc_kernel(const float* __restrict__ o_ln,
                          const float* __restrict__ fcwT,
                          const float* __restrict__ fcb,
                          float* __restrict__ out)
{
    __shared__ float At[16 * Ee];                 // 24 KB
    const int tid = threadIdx.x;
    const int nBase  = blockIdx.x * 16;
    const int coBase = blockIdx.y * 64;
    for (int i = tid; i < 16 * Ee; i += 128) At[i] = o_ln[(size_t)nBase * Ee + i];
    __syncthreads();

    const int wave = tid >> 5, lane = tid & 31, l16 = lane & 15, hi = lane >> 4;
    const int co0 = coBase + wave * 16;
    v8f c = {};
    for (int kc = 0; kc < Ee; kc += 48) {          // 8 chunks x 12 k-steps
        v2f af[12], bf[12];
#pragma unroll
        for (int t = 0; t < 12; t++) {
            int k = kc + t * 4;
            af[t].x = At[l16 * Ee + k + 2 * hi];
            af[t].y = At[l16 * Ee + k + 2 * hi + 1];
            bf[t].x = fcwT[(size_t)(k + 2 * hi) * Ee + co0 + l16];       // coalesced
            bf[t].y = fcwT[(size_t)(k + 2 * hi + 1) * Ee + co0 + l16];
        }
#pragma unroll
        for (int t = 0; t < 12; t++)
            c = __builtin_amdgcn_wmma_f32_16x16x4_f32(false, af[t], false, bf[t],
                                                      (short)0, c, false, false);
    }
#pragma unroll
    for (int r = 0; r < 8; r++) {
        int row = nBase + r + hi * 8;
        if (row < Nn) {
            int col = co0 + l16;
            out[(size_t)row * Ee + col] = c[r] + fcb[col];
        }
    }
}

// =====================================================================
// Launch. ws layout (floats):
//   Y[3*NT] | partial[1024] | stats[16] | qkv[3*48*SP*64] | fcwT[E*E]
// o_raw reuses Y[0] region, o_ln reuses Y[1..2] region. ~15.5 MB total.
// =====================================================================
extern "C" void kernel_launch(void* const* d_in, const int* in_sizes, int n_in,
                              void* d_out, int out_size, void* d_ws, size_t ws_size,
                              hipStream_t stream)
{
    const float* x   = (const float*)d_in[0];
    const float* wq  = (const float*)d_in[1];
    const float* wk  = (const float*)d_in[2];
    const float* wv  = (const float*)d_in[3];
    const float* qlw = (const float*)d_in[4];
    const float* qlb = (const float*)d_in[5];
    const float* klw = (const float*)d_in[6];
    const float* klb = (const float*)d_in[7];
    const float* vlw = (const float*)d_in[8];
    const float* vlb = (const float*)d_in[9];
    const float* olw = (const float*)d_in[10];
    const float* olb = (const float*)d_in[11];
    const float* fcw = (const float*)d_in[12];
    const float* fcb = (const float*)d_in[13];
    float* out = (float*)d_out;

    float* ws      = (float*)d_ws;
    float* Y       = ws;                       // 3*NT floats
    float* partial = Y + 3 * NT;               // 1024 (padded for alignment)
    float* stats   = partial + 1024;           // 16
    float* qkv     = stats + 16;               // 3*48*SP*64, 16B-aligned
    float* fcwT    = qkv + (size_t)3 * 48 * SP * Dd;  // E*E
    float* o_raw   = Y;                        // reuse (Y dead after qkv_build)
    float* o_ln    = Y + NT;                   // reuse, NP*Ee <= 2*NT

    adder_kernel<<<dim3(25, 6, 3), 256, 0, stream>>>(x, wq, wk, wv, Y, partial);
    stats_kernel<<<3, 256, 0, stream>>>(partial, stats);
    qkv_build_kernel<<<dim3(13, 48, 3), 256, 0, stream>>>(Y, stats,
                                                          qlw, qlb, klw, klb, vlw, vlb, qkv);
    wt_kernel<<<dim3(12, 12), 256, 0, stream>>>(fcw, fcwT);
    size_t attn_lds = (size_t)(2 * SP * Dd + 16 * Dd + 16 * SP + 128 + 32) * sizeof(float);
    attn_kernel<<<dim3(13, 48), 128, attn_lds, stream>>>(qkv, o_raw);
    oln_kernel<<<NP, 128, 0, stream>>>(o_raw, olw, olb, o_ln);
    fc_kernel<<<dim3(99, 6), 128, 0, stream>>>(o_ln, fcwT, fcb, out);
}